// YoloXWrapper_72430328479828
// MI455X (gfx1250) — compile-verified
//
#include <hip/hip_runtime.h>
#include <hip/hip_bf16.h>
#include <stdint.h>

#define B_IMG      32
#define N_ANCH     8400
#define N_CLS      80
#define ROW_F      85          // cx,cy,w,h,obj + 80 cls
#define K_PRE      512
#define MAX_PER    100
#define CONF_TH    0.25f
#define IOU_TH     0.45f
#define CLS_OFF    4096.0f
#define SORT_M     16384       // next pow2 >= 8400
#define TILE_ROWS  96          // anchors per decode block

typedef int v4i __attribute__((ext_vector_type(4)));
typedef __attribute__((address_space(1))) int  gint;    // global addrspace
typedef __attribute__((address_space(3))) int  lint;    // LDS addrspace
typedef __attribute__((address_space(1))) v4i  gv4i;
typedef __attribute__((address_space(3))) v4i  lv4i;

// ---------- CDNA5 async-copy path (guarded; falls back to sync loads) ------
#if defined(__AMDGCN__) && __has_builtin(__builtin_amdgcn_global_load_async_to_lds_b32)
#define HAVE_ASYNC_LDS 1
#else
#define HAVE_ASYNC_LDS 0
#endif
#if defined(__AMDGCN__) && __has_builtin(__builtin_amdgcn_global_load_async_to_lds_b128)
#define HAVE_ASYNC_LDS_B128 1
#else
#define HAVE_ASYNC_LDS_B128 0
#endif

__device__ __forceinline__ void wait_async_cnt0() {
#if defined(__AMDGCN__)
#if __has_builtin(__builtin_amdgcn_s_wait_asynccnt)
  __builtin_amdgcn_s_wait_asynccnt(0);
#else
  asm volatile("s_wait_asynccnt 0" ::: "memory");
#endif
#endif
}

// monotone map: float -> uint32 so that unsigned ascending == float ascending
__device__ __forceinline__ unsigned mono_f32(float f) {
  unsigned u = __float_as_uint(f);
  return (u & 0x80000000u) ? ~u : (u | 0x80000000u);
}

// ============================================================================
// Kernel A: stream pred, compute score = obj * max(cls) per anchor.
// One tile of TILE_ROWS anchors staged in LDS via CDNA5 async global->LDS copy
// (b128 granularity: 512B per wave-instruction).
// ============================================================================
__global__ __launch_bounds__(256)
void yolox_score_kernel(const float* __restrict__ pred, float* __restrict__ scores) {
  __shared__ __align__(16) float tile[TILE_ROWS * ROW_F];
  const int b  = blockIdx.y;
  const int a0 = blockIdx.x * TILE_ROWS;
  const int rows = min(TILE_ROWS, N_ANCH - a0);
  const int cnt  = rows * ROW_F;
  const float* g = pred + ((size_t)b * N_ANCH + a0) * ROW_F;

#if HAVE_ASYNC_LDS
  {
    float* gg = const_cast<float*>(g);
#if HAVE_ASYNC_LDS_B128
    const int cnt4 = cnt >> 2;             // tile base & image stride are 16B-aligned
    for (int t = threadIdx.x; t < cnt4; t += 256) {
      __builtin_amdgcn_global_load_async_to_lds_b128(
          (gv4i*)(gg + 4 * t), (lv4i*)(tile + 4 * t), 0, 0);
    }
    for (int t = (cnt4 << 2) + threadIdx.x; t < cnt; t += 256) {  // tail (normally empty)
      __builtin_amdgcn_global_load_async_to_lds_b32(
          (gint*)(gg + t), (lint*)(tile + t), 0, 0);
    }
#else
    for (int t = threadIdx.x; t < cnt; t += 256) {
      __builtin_amdgcn_global_load_async_to_lds_b32(
          (gint*)(gg + t), (lint*)(tile + t), 0, 0);
    }
#endif
    wait_async_cnt0();   // per-wave ASYNCcnt drain before LDS read
  }
#else
  for (int t = threadIdx.x; t < cnt; t += 256) tile[t] = g[t];
#endif

  // prefetch next tile while this one is consumed (emits global_prefetch_b8)
  if (a0 + TILE_ROWS < N_ANCH && threadIdx.x == 0)
    __builtin_prefetch(g + (size_t)TILE_ROWS * ROW_F, 0, 1);

  __syncthreads();

  if (threadIdx.x < rows) {
    const float* row = tile + threadIdx.x * ROW_F;
    const float obj = row[4];
    float m = row[5];
#pragma unroll 8
    for (int c = 6; c < 5 + N_CLS; ++c) m = fmaxf(m, row[c]);
    scores[(size_t)b * N_ANCH + a0 + threadIdx.x] = obj * m;
  }
}

// ============================================================================
// Kernel B: per image — max-reduce for conf, mask, exact top-512 sorted via
// full bitonic sort of 16384 x 64-bit keys in 128 KB of LDS (CDNA5 320KB WGP).
// key = (~mono(masked_score) << 32) | index  -> ascending sort == desc score,
// asc index on ties (matches jax.lax.top_k). Padding/masked sink to bottom.
// ============================================================================
__global__ __launch_bounds__(1024)
void yolox_topk_kernel(const float* __restrict__ scores,
                       int* __restrict__ top_idx, int* __restrict__ top_valid) {
  __shared__ unsigned long long keys[SORT_M];   // 128 KB
  __shared__ float sconf;
  const int b = blockIdx.x, tid = threadIdx.x;
  const float* s = scores + (size_t)b * N_ANCH;

  // image max (reuse keys[] as float scratch)
  float lm = -1.0f;
  for (int n = tid; n < N_ANCH; n += 1024) lm = fmaxf(lm, s[n]);
  float* red = (float*)keys;
  red[tid] = lm;
  __syncthreads();
  for (int off = 512; off > 0; off >>= 1) {
    if (tid < off) red[tid] = fmaxf(red[tid], red[tid + off]);
    __syncthreads();
  }
  if (tid == 0) sconf = fminf(CONF_TH, red[0]);
  __syncthreads();
  const float conf = sconf;
  __syncthreads();

  const float ninf = __int_as_float(0xFF800000);
  for (int n = tid; n < SORT_M; n += 1024) {
    unsigned long long key;
    if (n < N_ANCH) {
      const float v = s[n];
      const float mv = (v >= conf) ? v : ninf;
      key = ((unsigned long long)(~mono_f32(mv)) << 32) | (unsigned)n;
    } else {
      key = ~0ull;
    }
    keys[n] = key;
  }
  __syncthreads();

  // bitonic sort, ascending: 1024 threads x 8 pairs/step
  for (unsigned k = 2; k <= SORT_M; k <<= 1) {
    for (unsigned j = k >> 1; j > 0; j >>= 1) {
      for (unsigned p = tid; p < SORT_M / 2; p += 1024) {
        const unsigned i = 2u * p - (p & (j - 1u));
        const unsigned x = i ^ j;
        const unsigned long long a = keys[i], c = keys[x];
        const bool up = ((i & k) == 0);
        if ((a > c) == up) { keys[i] = c; keys[x] = a; }
      }
      __syncthreads();
    }
  }

  if (tid < K_PRE) {
    const unsigned long long key = keys[tid];
    const unsigned hi = (unsigned)(key >> 32);
    top_idx[b * K_PRE + tid]   = (int)(key & 0xFFFFFFFFu);
    top_valid[b * K_PRE + tid] = (hi & 0x80000000u) ? 0 : 1; // finite <=> nonneg score
  }
}

// ============================================================================
// Kernel C: gather 512 selected anchors (decode box, class max/argmax), then
// wave32 greedy NMS (one wave/image, keep bits in registers, __ballot for the
// cross-lane any()), stable compaction, write dets[32,100,7] + mask[32,100].
// ============================================================================
__global__ __launch_bounds__(512)
void yolox_nms_kernel(const float* __restrict__ pred,
                      const int* __restrict__ top_idx,
                      const int* __restrict__ top_valid,
                      float* __restrict__ dets, float* __restrict__ maskout) {
  __shared__ float bx1[K_PRE], by1[K_PRE], bx2[K_PRE], by2[K_PRE];
  __shared__ float bobj[K_PRE], bcc[K_PRE], bcp[K_PRE];
  __shared__ int   bval[K_PRE];
  __shared__ unsigned keepmask[32];
  __shared__ int   sel[MAX_PER];
  __shared__ int   scnt;
  const int b = blockIdx.x, t = threadIdx.x;

  { // gather + decode
    const int idx = top_idx[b * K_PRE + t];
    const int v   = top_valid[b * K_PRE + t];
    float x1 = 0.f, y1 = 0.f, x2 = 0.f, y2 = 0.f, obj = 0.f, cc = 0.f;
    int cp = 0;
    if (v) {
      const float* row = pred + ((size_t)b * N_ANCH + idx) * ROW_F;
      const float cx = row[0], cy = row[1], hw = row[2] * 0.5f, hh = row[3] * 0.5f;
      x1 = cx - hw; y1 = cy - hh; x2 = cx + hw; y2 = cy + hh;
      obj = row[4];
      cc = row[5];
#pragma unroll 8
      for (int c = 1; c < N_CLS; ++c) {
        const float vv = row[5 + c];
        if (vv > cc) { cc = vv; cp = c; }
      }
    }
    bx1[t] = x1; by1[t] = y1; bx2[t] = x2; by2[t] = y2;
    bobj[t] = obj; bcc[t] = cc; bcp[t] = (float)cp; bval[t] = v;
  }
  __syncthreads();

  if (t < 32) { // wave 0: sequential greedy NMS, zero barriers
    float rx1[16], ry1[16], rx2[16], ry2[16], rar[16];
#pragma unroll
    for (int k = 0; k < 16; ++k) {
      const int j = t + 32 * k;
      const float o = bcp[j] * CLS_OFF;
      rx1[k] = bx1[j] + o; ry1[k] = by1[j] + o;
      rx2[k] = bx2[j] + o; ry2[k] = by2[j] + o;
      rar[k] = (bx2[j] - bx1[j]) * (by2[j] - by1[j]);
    }
    unsigned keep = 0u;
    for (int i = 0; i < K_PRE; ++i) {
      const float o   = bcp[i] * CLS_OFF;           // LDS broadcast reads
      const float ix1 = bx1[i] + o, iy1 = by1[i] + o;
      const float ix2 = bx2[i] + o, iy2 = by2[i] + o;
      const float ia  = (bx2[i] - bx1[i]) * (by2[i] - by1[i]);
      const int   vi  = bval[i];
      int sup = 0;
#pragma unroll
      for (int k = 0; k < 16; ++k) {
        const float xx1 = fmaxf(ix1, rx1[k]), yy1 = fmaxf(iy1, ry1[k]);
        const float xx2 = fminf(ix2, rx2[k]), yy2 = fminf(iy2, ry2[k]);
        const float inter = fmaxf(xx2 - xx1, 0.f) * fmaxf(yy2 - yy1, 0.f);
        const float uni   = ia + rar[k] - inter;
        const int over = inter > IOU_TH * (uni + 1e-9f); // iou>T without div
        sup |= over & (int)((keep >> k) & 1u);
      }
      const int any = (__ballot(sup) != 0);
      if (t == (i & 31)) {
        const unsigned bit = (vi && !any) ? 1u : 0u;
        keep |= bit << (i >> 5);
      }
    }
    keepmask[t] = keep;
  }
  __syncthreads();

  if (t == 0) { // stable compaction (kept indices ascending == score order)
    int c = 0;
    for (int i = 0; i < K_PRE; ++i)
      if ((keepmask[i & 31] >> (i >> 5)) & 1u) { if (c < MAX_PER) sel[c] = i; ++c; }
    scnt = c;
  }
  __syncthreads();

  if (t < MAX_PER) {
    float* dp = dets + ((size_t)b * MAX_PER + t) * 7;
    if (t < scnt) {
      const int j = sel[t];
      dp[0] = bx1[j]; dp[1] = by1[j]; dp[2] = bx2[j]; dp[3] = by2[j];
      dp[4] = bobj[j]; dp[5] = bcc[j]; dp[6] = bcp[j];
      maskout[b * MAX_PER + t] = 1.0f;
    } else {
#pragma unroll
      for (int q = 0; q < 7; ++q) dp[q] = 0.f;
      maskout[b * MAX_PER + t] = 0.0f;
    }
  }
}

// ============================================================================
extern "C" void kernel_launch(void* const* d_in, const int* in_sizes, int n_in,
                              void* d_out, int out_size, void* d_ws, size_t ws_size,
                              hipStream_t stream) {
  const float* pred = (const float*)d_in[0];

  // workspace: scores[32*8400] f32 | top_idx[32*512] i32 | top_valid[32*512] i32
  float* scores   = (float*)d_ws;
  int*   top_idx  = (int*)(scores + (size_t)B_IMG * N_ANCH);
  int*   top_val  = top_idx + B_IMG * K_PRE;

  float* dets    = (float*)d_out;                       // [32,100,7]
  float* maskout = dets + (size_t)B_IMG * MAX_PER * 7;  // [32,100]

  const int tiles = (N_ANCH + TILE_ROWS - 1) / TILE_ROWS;  // 88
  yolox_score_kernel<<<dim3(tiles, B_IMG), 256, 0, stream>>>(pred, scores);
  yolox_topk_kernel<<<B_IMG, 1024, 0, stream>>>(scores, top_idx, top_val);
  yolox_nms_kernel<<<B_IMG, 512, 0, stream>>>(pred, top_idx, top_val, dets, maskout);
}